// Differentiable_Voxelizer_44693429682239
// MI455X (gfx1250) — compile-verified
//
#include <hip/hip_runtime.h>
#include <math.h>

// ---------------------------------------------------------------------------
// Differentiable voxelizer for gfx1250 (MI455X).
// Hot loop (winding number, ~1e9 point-vertex pairs) is recast as two skinny
// K=3 GEMMs per 16x16 tile and run on V_WMMA_F32_16X16X4_F32.
// Vertex data is packed into 32B records {na.xyz, na.v, v.xyz, |v|^2} so the
// inner loop is two b128 loads + register cndmask selects (no divergence).
// ---------------------------------------------------------------------------

typedef float v2f __attribute__((ext_vector_type(2)));
typedef float v8f __attribute__((ext_vector_type(8)));

#define HDRF      64                  // header floats at start of workspace
#define NMAX      65                  // max grid points per axis (BBOX_DENSITY+1)
#define MMAX      (NMAX*NMAX*NMAX)    // worst-case grid point count
#define WIND_EPS  1e-6f

// header layout (float* wsf / int* wsi aliased):
// wsf[0..2]=bmin  wsf[3..5]=bmax  wsf[20..22]=axis step
// wsi[6..8]=n(grid dims)  wsi[9]=M  wsi[10]=ntiles
// wsi[11..13]=bidx lo  wsi[14..16]=bidx hi  wsi[17..19]=box out sizes

__device__ __forceinline__ float dot3(const float a[3], const float b[3]) {
  return a[0]*b[0] + a[1]*b[1] + a[2]*b[2];
}

__device__ __forceinline__ void grid_point(int m, int Mtot, int n1, int n2,
                                           float bx, float by, float bz,
                                           float ax, float ay, float az,
                                           float p[3]) {
  if (m >= Mtot) m = Mtot - 1;          // clamp (padded rows of last tile)
  int k = m % n2;
  int q = m / n2;
  int j = q % n1;
  int i = q / n1;
  p[0] = bx + (float)i * ax;
  p[1] = by + (float)j * ay;
  p[2] = bz + (float)k * az;
}

// ---------------------------------------------------------------------------
// Kernel 1: bbox reduction + grid/layout parameters (single block).
// ---------------------------------------------------------------------------
__global__ void k_prep(const float* __restrict__ verts, int V,
                       const int* __restrict__ resPtr, float* __restrict__ wsf) {
  __shared__ float smin[3 * 256];
  __shared__ float smax[3 * 256];
  int t = threadIdx.x;
  float mn[3] = {1e30f, 1e30f, 1e30f};
  float mx[3] = {-1e30f, -1e30f, -1e30f};
  for (int v = t; v < V; v += 256) {
    for (int c = 0; c < 3; ++c) {
      float x = verts[v * 3 + c];
      mn[c] = fminf(mn[c], x);
      mx[c] = fmaxf(mx[c], x);
    }
  }
  for (int c = 0; c < 3; ++c) { smin[c * 256 + t] = mn[c]; smax[c * 256 + t] = mx[c]; }
  __syncthreads();
  for (int s = 128; s > 0; s >>= 1) {
    if (t < s) {
      for (int c = 0; c < 3; ++c) {
        smin[c * 256 + t] = fminf(smin[c * 256 + t], smin[c * 256 + t + s]);
        smax[c * 256 + t] = fmaxf(smax[c * 256 + t], smax[c * 256 + t + s]);
      }
    }
    __syncthreads();
  }
  if (t == 0) {
    int* wsi = (int*)wsf;
    float bmn[3], bmx[3], blen[3];
    float blmax = 0.0f;
    for (int c = 0; c < 3; ++c) {
      bmn[c] = smin[c * 256];
      bmx[c] = smax[c * 256];
      blen[c] = bmx[c] - bmn[c];
      blmax = fmaxf(blmax, blen[c]);
      wsf[c] = bmn[c];
      wsf[3 + c] = bmx[c];
    }
    float step_len = blmax / 64.0f;   // BBOX_DENSITY
    int n[3];
    for (int c = 0; c < 3; ++c) {
      n[c] = (step_len > 0.0f) ? ((int)(blen[c] / step_len) + 1) : 1;
      if (n[c] < 1) n[c] = 1;
      if (n[c] > NMAX) n[c] = NMAX;
      wsi[6 + c] = n[c];
      wsf[20 + c] = (n[c] > 1) ? (blen[c] / (float)(n[c] - 1)) : 0.0f;
    }
    int M = n[0] * n[1] * n[2];
    wsi[9] = M;
    wsi[10] = (M + 15) >> 4;
    int R = *resPtr;
    for (int c = 0; c < 3; ++c) {
      int b0 = (int)floorf((bmn[c] + 1.0f) * (float)R * 0.5f);
      int b1 = (int)floorf((bmx[c] + 1.0f) * (float)R * 0.5f);
      wsi[11 + c] = b0;
      wsi[14 + c] = b1;
      wsi[17 + c] = b1 - b0 + 1;
    }
  }
}

// ---------------------------------------------------------------------------
// Kernel 2: per-face geometry (cross product + angle-weighted corner weights).
// facedat[f*6 + 0..2] = cross, [3..5] = corner weight * area  (deterministic,
// no atomics; vertices gather in kernel 3).
// ---------------------------------------------------------------------------
__global__ void k_face(const float* __restrict__ verts,
                       const long long* __restrict__ faces, int F,
                       float* __restrict__ facedat) {
  int f = blockIdx.x * blockDim.x + threadIdx.x;
  if (f >= F) return;
  int i0 = (int)faces[f * 3 + 0];
  int i1 = (int)faces[f * 3 + 1];
  int i2 = (int)faces[f * 3 + 2];
  float v0[3], v1[3], v2[3];
  for (int c = 0; c < 3; ++c) {
    v0[c] = verts[i0 * 3 + c];
    v1[c] = verts[i1 * 3 + c];
    v2[c] = verts[i2 * 3 + c];
  }
  float A[3], B[3], C[3], D0[3], cr[3];
  for (int c = 0; c < 3; ++c) {
    A[c] = v1[c] - v0[c];
    B[c] = v2[c] - v1[c];
    C[c] = v0[c] - v2[c];
    D0[c] = v2[c] - v0[c];
  }
  cr[0] = A[1] * D0[2] - A[2] * D0[1];
  cr[1] = A[2] * D0[0] - A[0] * D0[2];
  cr[2] = A[0] * D0[1] - A[1] * D0[0];
  float area = 0.5f * __builtin_amdgcn_sqrtf(dot3(cr, cr));
  float nA = __builtin_amdgcn_sqrtf(dot3(A, A));
  float nB = __builtin_amdgcn_sqrtf(dot3(B, B));
  float nC = __builtin_amdgcn_sqrtf(dot3(C, C));
  float d0 = fminf(fmaxf(-dot3(A, C) / (1e-10f + nA * nC), -1.0f), 1.0f);
  float d1 = fminf(fmaxf(-dot3(A, B) / (1e-10f + nA * nB), -1.0f), 1.0f);
  float d2 = fminf(fmaxf(-dot3(B, C) / (1e-10f + nB * nC), -1.0f), 1.0f);
  // sin(2*acos(x)) = 2*x*sqrt(1-x^2)  (avoids acos entirely)
  float s20 = 2.0f * d0 * __builtin_amdgcn_sqrtf(fmaxf(0.0f, 1.0f - d0 * d0));
  float s21 = 2.0f * d1 * __builtin_amdgcn_sqrtf(fmaxf(0.0f, 1.0f - d1 * d1));
  float s22 = 2.0f * d2 * __builtin_amdgcn_sqrtf(fmaxf(0.0f, 1.0f - d2 * d2));
  float inv = 1.0f / (s20 + s21 + s22 + 1e-8f);
  float w0 = s20 * inv, w1 = s21 * inv, w2 = s22 * inv;
  // w = (w[:, [2,0,1]] + w[:, [1,2,0]]) * 0.5
  float c0 = 0.5f * (w2 + w1);
  float c1 = 0.5f * (w0 + w2);
  float c2 = 0.5f * (w1 + w0);
  facedat[f * 6 + 0] = cr[0];
  facedat[f * 6 + 1] = cr[1];
  facedat[f * 6 + 2] = cr[2];
  facedat[f * 6 + 3] = c0 * area;
  facedat[f * 6 + 4] = c1 * area;
  facedat[f * 6 + 5] = c2 * area;
}

// ---------------------------------------------------------------------------
// Kernel 3: per-vertex gather of normals/dual areas, then finalize into a
// packed 32B record per vertex:
//   vdat[v*8 + 0..3] = {na.x, na.y, na.z, na.v}
//   vdat[v*8 + 4..7] = {v.x, v.y, v.z, |v|^2}
// Face arrays (~390KB) stay L2-resident across all vertex threads.
// ---------------------------------------------------------------------------
__global__ void k_vert(const float* __restrict__ verts,
                       const long long* __restrict__ faces,
                       const float* __restrict__ facedat, int F, int V,
                       float* __restrict__ vdat) {
  int v = blockIdx.x * blockDim.x + threadIdx.x;
  if (v >= V) return;
  float vn[3] = {0.0f, 0.0f, 0.0f};
  float dual = 0.0f;
  for (int f = 0; f < F; ++f) {
    int i0 = (int)faces[f * 3 + 0];
    int i1 = (int)faces[f * 3 + 1];
    int i2 = (int)faces[f * 3 + 2];
    int m0 = (i0 == v), m1 = (i1 == v), m2 = (i2 == v);
    int mc = m0 + m1 + m2;
    if (mc) {
      float fm = (float)mc;
      vn[0] += fm * facedat[f * 6 + 0];
      vn[1] += fm * facedat[f * 6 + 1];
      vn[2] += fm * facedat[f * 6 + 2];
      if (m0) dual += facedat[f * 6 + 3];
      if (m1) dual += facedat[f * 6 + 4];
      if (m2) dual += facedat[f * 6 + 5];
    }
  }
  float nrm = __builtin_amdgcn_sqrtf(dot3(vn, vn));
  float s = dual / fmaxf(nrm, 1e-12f);
  float p[3] = {verts[v * 3 + 0], verts[v * 3 + 1], verts[v * 3 + 2]};
  float naL[3] = {vn[0] * s, vn[1] * s, vn[2] * s};
  float4 rec0, rec1;
  rec0.x = naL[0]; rec0.y = naL[1]; rec0.z = naL[2]; rec0.w = dot3(naL, p);
  rec1.x = p[0];   rec1.y = p[1];   rec1.z = p[2];   rec1.w = dot3(p, p);
  ((float4*)vdat)[v * 2 + 0] = rec0;
  ((float4*)vdat)[v * 2 + 1] = rec1;
}

// ---------------------------------------------------------------------------
// Kernel 4: winding number via WMMA. One wave = 16 grid points vs all verts.
//   G1 = P * naT   (p.na),   G2 = P * vT   (p.v)     [V_WMMA_F32_16X16X4_F32]
//   t  = |p|^2 + |v|^2 - 2*G2 ;  contrib = (cn - G1) / (t*sqrt(t) + eps)
// A layout: lane%16 = M row; VGPR{0,1} x lane-half = K 0..3 (K=3 padded w/ 0).
// B layout: lane%16 = N col; VGPR{0,1} x lane-half = K 0..3.
// D layout: VGPR r -> rows r (lanes 0-15) / r+8 (lanes 16-31), col = lane%16.
// ---------------------------------------------------------------------------
__global__ void k_wind(const float* __restrict__ wsf,
                       const float4* __restrict__ vdat,
                       float* __restrict__ vol, int V) {
  const int* wsi = (const int*)wsf;
  int lane = threadIdx.x & 31;
  int wave = blockIdx.x * (blockDim.x >> 5) + (threadIdx.x >> 5);
  int ntiles = wsi[10];
  if (wave >= ntiles) return;            // wave-uniform: EXEC stays all-ones

  int Mtot = wsi[9];
  int n1 = wsi[7], n2 = wsi[8];
  float bx = wsf[0], by = wsf[1], bz = wsf[2];
  float ax = wsf[20], ay = wsf[21], az = wsf[22];
  int m0 = wave << 4;
  bool lo = (lane < 16);

  // A operand: this lane's point row
  float pa[3];
  grid_point(m0 + (lane & 15), Mtot, n1, n2, bx, by, bz, ax, ay, az, pa);
  v2f Aop;
  Aop.x = lo ? pa[0] : pa[2];
  Aop.y = lo ? pa[1] : 0.0f;

  // |p|^2 for the 8 D-rows this lane owns
  int rbase = lo ? 0 : 8;
  float pp[8];
#pragma unroll
  for (int r = 0; r < 8; ++r) {
    float q[3];
    grid_point(m0 + rbase + r, Mtot, n1, n2, bx, by, bz, ax, ay, az, q);
    pp[r] = dot3(q, q);
  }

  float acc[8] = {0.f, 0.f, 0.f, 0.f, 0.f, 0.f, 0.f, 0.f};
  const v8f czero = {0.f, 0.f, 0.f, 0.f, 0.f, 0.f, 0.f, 0.f};
  int ntv = (V + 15) >> 4;
#pragma unroll 2
  for (int vt = 0; vt < ntv; ++vt) {
    int ncol = (vt << 4) + (lane & 15);
    bool valid = (ncol < V);
    int ncl = valid ? ncol : 0;
    // two b128 loads, uniform addressing, no divergence
    float4 rna = vdat[ncl * 2 + 0];   // na.xyz, na.v
    float4 rvx = vdat[ncl * 2 + 1];   // v.xyz, |v|^2
    v2f Bna, Bvx;
    Bna.x = lo ? rna.x : rna.z;
    Bna.y = lo ? rna.y : 0.0f;
    Bvx.x = lo ? rvx.x : rvx.z;
    Bvx.y = lo ? rvx.y : 0.0f;
    float cn_l = rna.w;
    float vv_l = rvx.w;
    if (!valid) { Bna.x = 0.0f; Bna.y = 0.0f; cn_l = 0.0f; }  // zero column

    v8f g1 = __builtin_amdgcn_wmma_f32_16x16x4_f32(
        false, Aop, false, Bna, (short)0, czero, false, false);
    v8f g2 = __builtin_amdgcn_wmma_f32_16x16x4_f32(
        false, Aop, false, Bvx, (short)0, czero, false, false);

#pragma unroll
    for (int r = 0; r < 8; ++r) {
      float t = fmaxf(pp[r] + vv_l - 2.0f * g2[r], 0.0f);
      float den = t * __builtin_amdgcn_sqrtf(t) + WIND_EPS;   // r^3 + eps
      acc[r] += (cn_l - g1[r]) * __builtin_amdgcn_rcpf(den);
    }
  }

  // reduce the 16 N-columns (within each half-wave group of 16 lanes)
#pragma unroll
  for (int r = 0; r < 8; ++r) {
#pragma unroll
    for (int off = 8; off > 0; off >>= 1)
      acc[r] += __shfl_xor(acc[r], off, 16);
  }

  if ((lane & 15) == 0) {
    const float inv4pi = 0.07957747154594767f;
#pragma unroll
    for (int r = 0; r < 8; ++r) {
      int m = m0 + rbase + r;
      if (m < Mtot) {
        float w = acc[r] * inv4pi;
        float z = (w - 0.5f) * 100.0f;
        vol[m] = 1.0f / (1.0f + __expf(-z));   // sigmoid
      }
    }
  }
}

// ---------------------------------------------------------------------------
// Kernel 5: trilinear upsample of vol into the bbox window of the R^3 output,
// zero elsewhere, with the final (2,1,0) transpose folded into indexing.
// out[z*R*R + y*R + x] = whole[x,y,z].
// ---------------------------------------------------------------------------
__device__ __forceinline__ int iclampi(int x, int lim) {
  return x < 0 ? 0 : (x >= lim ? lim - 1 : x);
}

__global__ void k_out(const float* __restrict__ wsf,
                      const float* __restrict__ vol,
                      float* __restrict__ out, int R, int total) {
  int t = blockIdx.x * blockDim.x + threadIdx.x;
  if (t >= total) return;
  const int* wsi = (const int*)wsf;
  int x = t % R;
  int y = (t / R) % R;
  int z = t / (R * R);
  int ox = x - wsi[11], oy = y - wsi[12], oz = z - wsi[13];
  int Ox = wsi[17], Oy = wsi[18], Oz = wsi[19];
  int n0 = wsi[6], n1 = wsi[7], n2 = wsi[8];
  float val = 0.0f;
  if (ox >= 0 && ox < Ox && oy >= 0 && oy < Oy && oz >= 0 && oz < Oz) {
    float cx = ((float)ox + 0.5f) * ((float)n0 / (float)Ox) - 0.5f;
    float cy = ((float)oy + 0.5f) * ((float)n1 / (float)Oy) - 0.5f;
    float cz = ((float)oz + 0.5f) * ((float)n2 / (float)Oz) - 0.5f;
    int ix = (int)floorf(cx), iy = (int)floorf(cy), iz = (int)floorf(cz);
    float fx = cx - (float)ix, fy = cy - (float)iy, fz = cz - (float)iz;
    int x0 = iclampi(ix, n0), x1 = iclampi(ix + 1, n0);
    int y0 = iclampi(iy, n1), y1 = iclampi(iy + 1, n1);
    int z0 = iclampi(iz, n2), z1 = iclampi(iz + 1, n2);
    int s0 = n1 * n2;
    float c000 = vol[x0 * s0 + y0 * n2 + z0];
    float c001 = vol[x0 * s0 + y0 * n2 + z1];
    float c010 = vol[x0 * s0 + y1 * n2 + z0];
    float c011 = vol[x0 * s0 + y1 * n2 + z1];
    float c100 = vol[x1 * s0 + y0 * n2 + z0];
    float c101 = vol[x1 * s0 + y0 * n2 + z1];
    float c110 = vol[x1 * s0 + y1 * n2 + z0];
    float c111 = vol[x1 * s0 + y1 * n2 + z1];
    float c00 = c000 + (c001 - c000) * fz;
    float c01 = c010 + (c011 - c010) * fz;
    float c10 = c100 + (c101 - c100) * fz;
    float c11 = c110 + (c111 - c110) * fz;
    float c0v = c00 + (c01 - c00) * fy;
    float c1v = c10 + (c11 - c10) * fy;
    val = c0v + (c1v - c0v) * fx;
  }
  out[t] = val;
}

// ---------------------------------------------------------------------------
extern "C" void kernel_launch(void* const* d_in, const int* in_sizes, int n_in,
                              void* d_out, int out_size, void* d_ws, size_t ws_size,
                              hipStream_t stream) {
  (void)n_in; (void)ws_size;
  const float* verts = (const float*)d_in[0];
  const long long* faces = (const long long*)d_in[1];
  const int* resPtr = (const int*)d_in[2];
  int V = in_sizes[0] / 3;
  int F = in_sizes[1] / 3;
  int R = (int)lround(cbrt((double)out_size));

  float* wsf = (float*)d_ws;
  size_t off_face = HDRF;                       // 6*F floats
  size_t off_vdat = off_face + (size_t)6 * F;   // 8*V floats (16B-aligned)
  size_t off_vol  = off_vdat + (size_t)8 * V;   // up to MMAX floats

  k_prep<<<1, 256, 0, stream>>>(verts, V, resPtr, wsf);
  k_face<<<(F + 255) / 256, 256, 0, stream>>>(verts, faces, F, wsf + off_face);
  k_vert<<<(V + 255) / 256, 256, 0, stream>>>(verts, faces, wsf + off_face, F, V,
                                              wsf + off_vdat);
  // worst-case tile count (grid dims are device-side); surplus waves early-out
  int maxTiles = (MMAX + 15) / 16;
  int blocks = (maxTiles + 7) / 8;              // 8 waves (of 32) per block
  k_wind<<<blocks, 256, 0, stream>>>(wsf, (const float4*)(wsf + off_vdat),
                                     wsf + off_vol, V);
  k_out<<<(out_size + 255) / 256, 256, 0, stream>>>(wsf, wsf + off_vol,
                                                    (float*)d_out, R, out_size);
}